// MultiHeadAttention_77232101917088
// MI455X (gfx1250) — compile-verified
//
#include <hip/hip_runtime.h>

typedef float v2f __attribute__((ext_vector_type(2)));
typedef float v8f __attribute__((ext_vector_type(8)));

#define S      2048
#define DM     2048
#define NH     16
#define HD     128
static constexpr float kScale = 0.08838834764831845f; // 1/sqrt(128)

// ---------- A: Q[b][n] = sum_k x[b, S-1, k] * Wq[k][n]   (both batches fused)
__global__ void mha_q_proj(const float* __restrict__ x,
                           const float* __restrict__ Wq,
                           float* __restrict__ Q) {
  int n = blockIdx.x * blockDim.x + threadIdx.x;
  const float* xl0 = x + ((size_t)(0 * S + (S - 1))) * DM;
  const float* xl1 = x + ((size_t)(1 * S + (S - 1))) * DM;
  float a0 = 0.f, a1 = 0.f;
  for (int k = 0; k < DM; ++k) {
    float w = Wq[(size_t)k * DM + n];
    a0 = fmaf(xl0[k], w, a0);
    a1 = fmaf(xl1[k], w, a1);
  }
  Q[n] = a0;
  Q[DM + n] = a1;
}

// ---------- B: U[b][d][h] = sum_e Wk[d][h*128+e] * Q[b][h*128+e]
// one wave32 per row d, shuffle reduction; both batches fused (Wk read once)
__global__ void mha_u_proj(const float* __restrict__ Wk,
                           const float* __restrict__ Q,
                           float* __restrict__ U) {
  int lane = threadIdx.x & 31;
  int warp = threadIdx.x >> 5;
  int d = blockIdx.x * 8 + warp;
  const float* wrow = Wk + (size_t)d * DM;
  for (int h = 0; h < NH; ++h) {
    float a0 = 0.f, a1 = 0.f;
    int base = h * HD;
    #pragma unroll
    for (int i = 0; i < 4; ++i) {
      int j = base + i * 32 + lane;
      float w = wrow[j];
      a0 = fmaf(w, Q[j], a0);
      a1 = fmaf(w, Q[DM + j], a1);
    }
    #pragma unroll
    for (int off = 16; off > 0; off >>= 1) {
      a0 += __shfl_xor(a0, off);
      a1 += __shfl_xor(a1, off);
    }
    if (lane == 0) {
      U[(size_t)d * NH + h]        = a0;   // batch 0
      U[(size_t)(DM + d) * NH + h] = a1;   // batch 1 (offset DM*NH)
    }
  }
}

// ---------- C: scores[b][h][j] = sum_d x[b][j][d] * U[b][d][h]
// GEMM M=2048(j) x N=16(h) x K=2048(d) via V_WMMA_F32_16X16X4_F32
// one wave per 16-row M tile; 8 waves/block; grid (16, 2)
__global__ void mha_scores_wmma(const float* __restrict__ x,
                                const float* __restrict__ U,
                                float* __restrict__ scores) {
  int b    = blockIdx.y;
  int wave = threadIdx.x >> 5;
  int lane = threadIdx.x & 31;
  int lo   = lane & 15;      // M row (A) / N col (B) within tile
  int hi   = lane >> 4;      // K-pair selector per ISA layout
  int mbase = (blockIdx.x * 8 + wave) * 16;

  const float* xa = x + ((size_t)b * S + (mbase + lo)) * DM; // this lane's A row
  const float* Ub = U + (size_t)b * DM * NH;

  v8f c = {};
  for (int k0 = 0; k0 < DM; k0 += 4) {
    int kk = k0 + 2 * hi;                       // A/B VGPR0 holds K={kk}, VGPR1 K={kk+1}
    v2f a = *(const v2f*)(xa + kk);             // A: x[row, kk], x[row, kk+1]
    v2f bm;
    bm.x = Ub[(size_t)kk * NH + lo];            // B: U[kk,   h=lo]
    bm.y = Ub[(size_t)(kk + 1) * NH + lo];      //    U[kk+1, h=lo]
    c = __builtin_amdgcn_wmma_f32_16x16x4_f32(false, a, false, bm,
                                              (short)0, c, false, false);
  }
  // D layout: VGPR r, lanes 0-15 -> M=r, N=lane; lanes 16-31 -> M=r+8, N=lane-16
  #pragma unroll
  for (int r = 0; r < 8; ++r) {
    int m = mbase + r + 8 * hi;                 // key position j
    scores[((size_t)b * NH + lo) * S + m] = c[r];
  }
}

// ---------- D: softmax over keys per (b,h); last-row causal mask is a no-op
__global__ void mha_softmax(float* __restrict__ attn) {
  float* row = attn + (size_t)blockIdx.x * S;
  __shared__ float red[256];
  int t = threadIdx.x;
  float m = -3.402823466e38f;
  for (int j = t; j < S; j += 256) m = fmaxf(m, row[j]);
  red[t] = m; __syncthreads();
  for (int off = 128; off > 0; off >>= 1) {
    if (t < off) red[t] = fmaxf(red[t], red[t + off]);
    __syncthreads();
  }
  m = red[0]; __syncthreads();
  float s = 0.f;
  for (int j = t; j < S; j += 256) {
    float e = expf((row[j] - m) * kScale);
    row[j] = e;
    s += e;
  }
  red[t] = s; __syncthreads();
  for (int off = 128; off > 0; off >>= 1) {
    if (t < off) red[t] += red[t + off];
    __syncthreads();
  }
  float inv = 1.f / red[0];
  for (int j = t; j < S; j += 256) row[j] *= inv;
}

// ---------- E: t[b][h][n] = sum_j attn[b][h][j] * x[b][j][n]
// GEMM M=16(h) x N=2048(n) x K=2048(j) via V_WMMA_F32_16X16X4_F32
__global__ void mha_attnx_wmma(const float* __restrict__ attn,
                               const float* __restrict__ x,
                               float* __restrict__ tOut) {
  int b    = blockIdx.y;
  int wave = threadIdx.x >> 5;
  int lane = threadIdx.x & 31;
  int lo   = lane & 15;
  int hi   = lane >> 4;
  int nbase = (blockIdx.x * 8 + wave) * 16;

  const float* arow = attn + ((size_t)b * NH + lo) * S;  // A row: head = lo
  const float* xb   = x + (size_t)b * S * DM;

  v8f c = {};
  for (int k0 = 0; k0 < S; k0 += 4) {
    int kk = k0 + 2 * hi;
    v2f a = *(const v2f*)(arow + kk);                    // attn[h, kk], attn[h, kk+1]
    v2f bm;
    bm.x = xb[(size_t)kk * DM + nbase + lo];             // x[kk,   nbase+lo]
    bm.y = xb[(size_t)(kk + 1) * DM + nbase + lo];       // x[kk+1, nbase+lo]
    c = __builtin_amdgcn_wmma_f32_16x16x4_f32(false, a, false, bm,
                                              (short)0, c, false, false);
  }
  #pragma unroll
  for (int r = 0; r < 8; ++r) {
    int h = r + 8 * hi;
    tOut[((size_t)b * NH + h) * DM + nbase + lo] = c[r];
  }
}

// ---------- F: ctx[b][h*128+e] = sum_d t[b][h][d] * Wv[d][h*128+e]
// one block per head; both batches fused (Wv read once)
__global__ void mha_ctx(const float* __restrict__ tIn,
                        const float* __restrict__ Wv,
                        float* __restrict__ ctx) {
  int h = blockIdx.x;
  int e = threadIdx.x; // 0..127
  const float* t0 = tIn + (size_t)h * DM;
  const float* t1 = tIn + (size_t)(NH + h) * DM;
  float a0 = 0.f, a1 = 0.f;
  for (int d = 0; d < DM; ++d) {
    float w = Wv[(size_t)d * DM + h * HD + e];
    a0 = fmaf(t0[d], w, a0);
    a1 = fmaf(t1[d], w, a1);
  }
  ctx[h * HD + e]      = a0;
  ctx[DM + h * HD + e] = a1;
}

// ---------- G: out[b][n] = sum_m ctx[b][m] * Wo[m][n] + bo[n]
__global__ void mha_out(const float* __restrict__ ctx,
                        const float* __restrict__ Wo,
                        const float* __restrict__ bo,
                        float* __restrict__ out) {
  int n = blockIdx.x * blockDim.x + threadIdx.x;
  float a0 = bo[n], a1 = bo[n];
  for (int m = 0; m < DM; ++m) {
    float w = Wo[(size_t)m * DM + n];
    a0 = fmaf(ctx[m],      w, a0);
    a1 = fmaf(ctx[DM + m], w, a1);
  }
  out[n]      = a0;
  out[DM + n] = a1;
}

extern "C" void kernel_launch(void* const* d_in, const int* in_sizes, int n_in,
                              void* d_out, int out_size, void* d_ws, size_t ws_size,
                              hipStream_t stream) {
  (void)in_sizes; (void)n_in; (void)out_size; (void)ws_size;
  const float* x  = (const float*)d_in[0];
  const float* Wq = (const float*)d_in[1];
  const float* Wk = (const float*)d_in[2];
  const float* Wv = (const float*)d_in[3];
  const float* Wo = (const float*)d_in[4];
  const float* bo = (const float*)d_in[5];
  float* out = (float*)d_out;

  // workspace layout (floats): Q[2*2048] | U[2*2048*16] | attn[2*16*2048] |
  //                            t[2*16*2048] | ctx[2*2048]   (~0.8 MB total)
  float* ws  = (float*)d_ws;
  float* Q    = ws;
  float* U    = Q    + 2 * DM;
  float* attn = U    + 2 * DM * NH;
  float* tB   = attn + 2 * NH * S;
  float* ctx  = tB   + 2 * NH * DM;

  mha_q_proj     <<<DM / 256, 256, 0, stream>>>(x, Wq, Q);
  mha_u_proj     <<<DM / 8,   256, 0, stream>>>(Wk, Q, U);
  mha_scores_wmma<<<dim3(16, 2), 256, 0, stream>>>(x, U, attn);
  mha_softmax    <<<2 * NH,   256, 0, stream>>>(attn);
  mha_attnx_wmma <<<dim3(16, 2), 256, 0, stream>>>(attn, x, tB);
  mha_ctx        <<<NH,       HD,  0, stream>>>(tB, Wv, ctx);
  mha_out        <<<DM / 256, 256, 0, stream>>>(ctx, Wo, bo, out);
}